// Block_41059887350054
// MI455X (gfx1250) — compile-verified
//
#include <hip/hip_runtime.h>

typedef __attribute__((ext_vector_type(2))) float v2f;
typedef __attribute__((ext_vector_type(8))) float v8f;

#define D 128

// ---------------------------------------------------------------------------
// 1) init: zero agg[N*128], set deg[i] = 1.0f (self-loop contributes 1 to deg)
// ---------------------------------------------------------------------------
__global__ void gcn_init(float4* __restrict__ agg4, float* __restrict__ deg, int N) {
    int t = blockIdx.x * blockDim.x + threadIdx.x;       // N*32 threads
    if (t < N * (D / 4)) agg4[t] = make_float4(0.f, 0.f, 0.f, 0.f);
    if (t < N) deg[t] = 1.0f;
}

// ---------------------------------------------------------------------------
// 2) degree accumulation over src indices (int64 edge list)
// ---------------------------------------------------------------------------
__global__ void gcn_deg_accum(const long long* __restrict__ src,
                              float* __restrict__ deg, int E) {
    int t = blockIdx.x * blockDim.x + threadIdx.x;
    if (t < E) atomicAdd(&deg[(int)src[t]], 1.0f);
}

// ---------------------------------------------------------------------------
// 3) dis = deg^-0.5 (in place)
// ---------------------------------------------------------------------------
__global__ void gcn_deg_rsqrt(float* __restrict__ deg, int N) {
    int t = blockIdx.x * blockDim.x + threadIdx.x;
    if (t < N) deg[t] = rsqrtf(deg[t]);
}

// ---------------------------------------------------------------------------
// 4) h = x @ W^T + b  via V_WMMA_F32_16X16X4_F32 (fp32 exact, wave32)
//    one wave -> one 16x16 tile of h; K loop 128/4 = 32 WMMAs
//    A (16x4): lanes 0-15 = M, vgpr{0,1} = K{0,1}; lanes 16-31 carry K{2,3}
//    B (4x16): same striping with lane -> N, sourced from W rows (W is [Dout,Din])
//    C/D: vgpr v = row M=v (lanes 0-15) / M=8+v (lanes 16-31), N = lane&15
// ---------------------------------------------------------------------------
__global__ __launch_bounds__(256) void gcn_gemm_wmma(
    const float* __restrict__ x,   // [N, 128]
    const float* __restrict__ W,   // [128, 128] row-major (D_OUT, D_IN)
    const float* __restrict__ b,   // [128]
    float* __restrict__ h,         // [N, 128]  (aliases d_out)
    int N) {
    const int lane = threadIdx.x & 31;
    const int wave = threadIdx.x >> 5;
    const int gw   = blockIdx.x * 8 + wave;   // global wave id
    const int tileN = gw & 7;                 // 128/16 = 8 column tiles
    const int tileM = gw >> 3;
    const int row0 = tileM * 16;
    if (row0 >= N) return;                    // wave-uniform guard (EXEC stays full)
    const int col0 = tileN * 16;
    const int ml = lane & 15;                 // M for A, N for B/C/D
    const int kh = (lane >> 4) << 1;          // K sub-offset: 0 or 2

    const float bias = b[col0 + ml];
    v8f acc;
#pragma unroll
    for (int i = 0; i < 8; ++i) acc[i] = bias;

    const float* xrow = x + (size_t)(row0 + ml) * D;
    const float* wrow = W + (size_t)(col0 + ml) * D;

#pragma unroll 4
    for (int k = 0; k < D; k += 4) {
        v2f a, bb;
        a.x  = xrow[k + kh];
        a.y  = xrow[k + kh + 1];
        bb.x = wrow[k + kh];
        bb.y = wrow[k + kh + 1];
        acc = __builtin_amdgcn_wmma_f32_16x16x4_f32(
            /*neg_a=*/false, a, /*neg_b=*/false, bb,
            /*c_mod=*/(short)0, acc, /*reuse_a=*/false, /*reuse_b=*/false);
    }

    const int mbase = row0 + ((lane >> 4) << 3);   // +8 for upper half-wave
#pragma unroll
    for (int v = 0; v < 8; ++v)
        h[(size_t)(mbase + v) * D + col0 + ml] = acc[v];
}

// ---------------------------------------------------------------------------
// 5) edge scatter: one wave per edge; lane covers 4 channels (32*4 = 128)
//    agg[dst] += dis[src]*dis[dst] * h[src]   (f32 atomics, L2-resident)
// ---------------------------------------------------------------------------
__global__ __launch_bounds__(256) void gcn_edge_scatter(
    const long long* __restrict__ src,
    const long long* __restrict__ dst,
    const float* __restrict__ dis,
    const float* __restrict__ h,
    float* __restrict__ agg, int E) {
    const int lane = threadIdx.x & 31;
    const int e = blockIdx.x * 8 + (threadIdx.x >> 5);
    if (e >= E) return;
    const int s = (int)src[e];
    const int d = (int)dst[e];
    const float norm = dis[s] * dis[d];
    const float4 hv = *(const float4*)(h + (size_t)s * D + lane * 4);
    float* ap = agg + (size_t)d * D + lane * 4;
    atomicAdd(ap + 0, norm * hv.x);
    atomicAdd(ap + 1, norm * hv.y);
    atomicAdd(ap + 2, norm * hv.z);
    atomicAdd(ap + 3, norm * hv.w);
}

// ---------------------------------------------------------------------------
// 6) finalize: add self-loop term h[i]*dis[i]^2, relu, scale; in-place on d_out
// ---------------------------------------------------------------------------
__global__ void gcn_finalize(const float* __restrict__ agg,
                             const float* __restrict__ dis,
                             float* __restrict__ h_out, int N) {
    int t = blockIdx.x * blockDim.x + threadIdx.x;   // N*32 threads
    if (t >= N * (D / 4)) return;
    const int i  = t >> 5;
    const int c4 = (t & 31) * 4;
    const float d2 = dis[i] * dis[i];
    const float coef = 0.125f;   // sqrt(C_SIGMA / D_OUT) = sqrt(2/128)
    float4 hv = *(const float4*)(h_out + (size_t)i * D + c4);
    float4 av = *(const float4*)(agg  + (size_t)i * D + c4);
    float4 o;
    o.x = coef * fmaxf(av.x + d2 * hv.x, 0.f);
    o.y = coef * fmaxf(av.y + d2 * hv.y, 0.f);
    o.z = coef * fmaxf(av.z + d2 * hv.z, 0.f);
    o.w = coef * fmaxf(av.w + d2 * hv.w, 0.f);
    *(float4*)(h_out + (size_t)i * D + c4) = o;
}

// ---------------------------------------------------------------------------
extern "C" void kernel_launch(void* const* d_in, const int* in_sizes, int n_in,
                              void* d_out, int out_size, void* d_ws, size_t ws_size,
                              hipStream_t stream) {
    const float*     x  = (const float*)d_in[0];            // [N,128] f32
    const long long* ei = (const long long*)d_in[1];        // [2,E]   i64
    const float*     W  = (const float*)d_in[2];            // [128,128] f32
    const float*     b  = (const float*)d_in[3];            // [128] f32
    float* out = (float*)d_out;                             // [N,128] f32; doubles as h

    const int N = in_sizes[0] / D;
    const int E = in_sizes[1] / 2;
    const long long* src = ei;
    const long long* dst = ei + E;

    float* agg = (float*)d_ws;                 // N*128 floats
    float* deg = agg + (size_t)N * D;          // N floats (becomes dis)

    // 1) zero agg, deg = 1 (self loop)
    {
        int threads = N * (D / 4);
        gcn_init<<<(threads + 255) / 256, 256, 0, stream>>>((float4*)agg, deg, N);
    }
    // 2) degree accumulation over src
    gcn_deg_accum<<<(E + 255) / 256, 256, 0, stream>>>(src, deg, E);
    // 3) dis = rsqrt(deg)
    gcn_deg_rsqrt<<<(N + 255) / 256, 256, 0, stream>>>(deg, N);
    // 4) GEMM h = x @ W^T + b  (h stored in d_out)
    {
        int waves  = (N / 16) * (D / 16);      // N%16==0 for N=100000
        int blocks = waves / 8;
        gcn_gemm_wmma<<<blocks, 256, 0, stream>>>(x, W, b, out, N);
    }
    // 5) edge gather-scale-scatter (8 edges per 256-thread block)
    gcn_edge_scatter<<<(E + 7) / 8, 256, 0, stream>>>(src, dst, deg, out, agg, E);
    // 6) finalize (in-place on d_out)
    {
        int threads = N * (D / 4);
        gcn_finalize<<<(threads + 255) / 256, 256, 0, stream>>>(agg, deg, out, N);
    }
}